// base_model_51548197486921
// MI455X (gfx1250) — compile-verified
//
#include <hip/hip_runtime.h>

typedef __attribute__((ext_vector_type(2))) float v2f;
typedef __attribute__((ext_vector_type(8))) float v8f;

#define B_  4
#define L_  200
#define N_  160
#define HS_ 768
#define HD_ 256
#define C_  16
#define LT_ 13   // ceil(L/16)

// ---------------------------------------------------------------------------
// Kernel 1: dense_h[b,l,:] = (l < length[b] ? transformer_out[b,l+1,:] : 0) @ Wd + bd
// One wave (32 threads) computes one 16x16 output tile via V_WMMA_F32_16X16X4_F32.
// f32 WMMA layouts (wave32):
//   A 16x4 : lanes 0-15 -> M=lane, {v0,v1} = K=0,1 ; lanes 16-31 -> M=lane-16, K=2,3
//   B 4x16 : lanes 0-15 -> N=lane, {v0,v1} = K=0,1 ; lanes 16-31 -> N=lane-16, K=2,3
//   C/D    : vgpr r: lanes 0-15 -> (M=r, N=lane); lanes 16-31 -> (M=r+8, N=lane-16)
// ---------------------------------------------------------------------------
__global__ __launch_bounds__(32) void k_dense(const float* __restrict__ TO,
                                              const float* __restrict__ Wd,
                                              const float* __restrict__ bd,
                                              const int*   __restrict__ length,
                                              float*       __restrict__ dense_h) {
    int blk = blockIdx.x;
    int b   = blk / (LT_ * (HD_ / 16));
    int rem = blk % (LT_ * (HD_ / 16));
    int lt  = rem / (HD_ / 16);
    int ht  = rem % (HD_ / 16);

    int lane = threadIdx.x;
    int half = lane >> 4;      // 0: K=0,1   1: K=2,3
    int l16  = lane & 15;

    int rowA = lt * 16 + l16;
    int len  = length[b];
    bool vrow = (rowA < L_) && (rowA < len);
    float scale = vrow ? 1.0f : 0.0f;
    const float* ap = TO + ((size_t)b * (L_ + 1) + 1 + (vrow ? rowA : 0)) * (size_t)HS_;

    int col = ht * 16 + l16;
    int koff = half * 2;

    v8f acc = {};
    for (int kb = 0; kb < HS_; kb += 4) {
        v2f a, bm;
        a.x  = ap[kb + koff]     * scale;
        a.y  = ap[kb + koff + 1] * scale;
        bm.x = Wd[(size_t)(kb + koff)     * HD_ + col];
        bm.y = Wd[(size_t)(kb + koff + 1) * HD_ + col];
        acc = __builtin_amdgcn_wmma_f32_16x16x4_f32(false, a, false, bm,
                                                    (short)0, acc, false, false);
    }

    float bias = bd[col];
#pragma unroll
    for (int r = 0; r < 8; ++r) {
        int row = lt * 16 + r + half * 8;
        if (row < L_)
            dense_h[((size_t)b * L_ + row) * HD_ + col] = acc[r] + bias;
    }
}

// ---------------------------------------------------------------------------
// Kernel 2: segmented mean-pool over sorted word_mapback.
// One block per batch, thread d owns channel d; groups are contiguous (sorted).
// ---------------------------------------------------------------------------
__global__ __launch_bounds__(HD_) void k_pool(const float* __restrict__ dense_h,
                                              const int*   __restrict__ wmap,
                                              const int*   __restrict__ length,
                                              float*       __restrict__ pooled) {
    int b = blockIdx.x;
    int d = threadIdx.x;

    for (int n = 0; n < N_; ++n)
        pooled[((size_t)b * N_ + n) * HD_ + d] = 0.0f;

    int len = length[b];
    if (len > L_) len = L_;

    int   cur = -1;
    float acc = 0.0f;
    int   cnt = 0;
    for (int l = 0; l < len; ++l) {
        int n = wmap[b * L_ + l];      // uniform across the block -> scalar load
        if (n != cur) {
            if (cur >= 0)
                pooled[((size_t)b * N_ + cur) * HD_ + d] = acc / (float)cnt;
            cur = n; acc = 0.0f; cnt = 0;
        }
        acc += dense_h[((size_t)b * L_ + l) * HD_ + d];
        ++cnt;
    }
    if (cur >= 0)
        pooled[((size_t)b * N_ + cur) * HD_ + d] = acc / (float)cnt;
}

// ---------------------------------------------------------------------------
// Kernel 3: A1 = pooled @ Wc[0:HD], A2 = pooled @ Wc[HD:2HD],
//           T  = (pooled * tok_mask) @ Wc[2HD:3HD]     (f32 WMMA, C_=16 = one tile)
// grid = B * (N/16) * 3 waves.
// ---------------------------------------------------------------------------
__global__ __launch_bounds__(32) void k_proj(const float* __restrict__ pooled,
                                             const float* __restrict__ Wc,
                                             const int*   __restrict__ token_length,
                                             float* __restrict__ A1,
                                             float* __restrict__ A2,
                                             float* __restrict__ T) {
    int blk   = blockIdx.x;
    int b     = blk / ((N_ / 16) * 3);
    int rem   = blk % ((N_ / 16) * 3);
    int nt    = rem / 3;
    int which = rem % 3;

    int lane = threadIdx.x;
    int half = lane >> 4;
    int l16  = lane & 15;

    int rowA = nt * 16 + l16;
    int tokl = token_length[b];
    float scale = (which == 2 && rowA >= tokl) ? 0.0f : 1.0f;

    const float* ap = pooled + ((size_t)b * N_ + rowA) * HD_;
    const float* wp = Wc + (size_t)which * HD_ * C_;
    int koff = half * 2;

    v8f acc = {};
    for (int kb = 0; kb < HD_; kb += 4) {
        v2f a, bm;
        a.x  = ap[kb + koff]     * scale;
        a.y  = ap[kb + koff + 1] * scale;
        bm.x = wp[(size_t)(kb + koff)     * C_ + l16];
        bm.y = wp[(size_t)(kb + koff + 1) * C_ + l16];
        acc = __builtin_amdgcn_wmma_f32_16x16x4_f32(false, a, false, bm,
                                                    (short)0, acc, false, false);
    }

    float* out = (which == 0) ? A1 : ((which == 1) ? A2 : T);
#pragma unroll
    for (int r = 0; r < 8; ++r) {
        int n = nt * 16 + r + half * 8;
        out[((size_t)b * N_ + n) * C_ + l16] = acc[r];
    }
}

// ---------------------------------------------------------------------------
// Kernel 4: S[b,n,c] = cumsum_n T[b,n,c]   (tiny; one thread per (b,c))
// ---------------------------------------------------------------------------
__global__ __launch_bounds__(B_ * C_) void k_scan(const float* __restrict__ T,
                                                  float* __restrict__ S) {
    int t = threadIdx.x;
    int b = t / C_;
    int c = t % C_;
    float run = 0.0f;
    for (int n = 0; n < N_; ++n) {
        run += T[((size_t)b * N_ + n) * C_ + c];
        S[((size_t)b * N_ + n) * C_ + c] = run;
    }
}

// ---------------------------------------------------------------------------
// Kernel 5: fused epilogue, streaming the [B,N,N,C] output.
// ---------------------------------------------------------------------------
__global__ __launch_bounds__(256) void k_emit(const float* __restrict__ A1,
                                              const float* __restrict__ A2,
                                              const float* __restrict__ S,
                                              const float* __restrict__ bc,
                                              const int*   __restrict__ token_length,
                                              float* __restrict__ out) {
    long idx = (long)blockIdx.x * 256 + threadIdx.x;
    int  c   = (int)(idx & (C_ - 1));
    long t   = idx >> 4;
    int  k   = (int)(t % N_);
    long t2  = t / N_;
    int  i   = (int)(t2 % N_);
    int  b   = (int)(t2 / N_);

    int tokl = token_length[b];
    float v = 0.0f;
    if (i <= k && i < tokl && k < tokl) {
        float sp   = (i > 0) ? S[((size_t)b * N_ + (i - 1)) * C_ + c] : 0.0f;
        float span = (S[((size_t)b * N_ + k) * C_ + c] - sp) / (float)(k - i + 1);
        v = A1[((size_t)b * N_ + i) * C_ + c]
          + A2[((size_t)b * N_ + k) * C_ + c]
          + span + bc[c];
    }
    out[idx] = v;
}

// ---------------------------------------------------------------------------
extern "C" void kernel_launch(void* const* d_in, const int* in_sizes, int n_in,
                              void* d_out, int out_size, void* d_ws, size_t ws_size,
                              hipStream_t stream) {
    const float* TO   = (const float*)d_in[0];   // [B, L+1, HS]
    const float* Wd   = (const float*)d_in[1];   // [HS, HD]
    const float* bd   = (const float*)d_in[2];   // [HD]
    const float* Wc   = (const float*)d_in[3];   // [3*HD, C]
    const float* bc   = (const float*)d_in[4];   // [C]
    const int*   wmap = (const int*)d_in[5];     // [B, L]
    const int*   tokl = (const int*)d_in[6];     // [B]
    const int*   len  = (const int*)d_in[7];     // [B]

    float* ws      = (float*)d_ws;
    float* dense_h = ws;                                   // B*L*HD   = 204800
    float* pooled  = dense_h + (size_t)B_ * L_ * HD_;      // B*N*HD   = 163840
    float* A1      = pooled  + (size_t)B_ * N_ * HD_;      // B*N*C    = 10240
    float* A2      = A1      + (size_t)B_ * N_ * C_;
    float* T       = A2      + (size_t)B_ * N_ * C_;
    float* S       = T       + (size_t)B_ * N_ * C_;

    float* out = (float*)d_out;                            // [B, N, N, C]

    k_dense<<<B_ * LT_ * (HD_ / 16), 32, 0, stream>>>(TO, Wd, bd, len, dense_h);
    k_pool <<<B_, HD_, 0, stream>>>(dense_h, wmap, len, pooled);
    k_proj <<<B_ * (N_ / 16) * 3, 32, 0, stream>>>(pooled, Wc, tokl, A1, A2, T);
    k_scan <<<1, B_ * C_, 0, stream>>>(T, S);
    k_emit <<<(B_ * N_ * N_ * C_) / 256, 256, 0, stream>>>(A1, A2, S, bc, tokl, out);
}